// ProbabilisticSwitch_52046413693048
// MI455X (gfx1250) — compile-verified
//
#include <hip/hip_runtime.h>
#include <hip/hip_bf16.h>

// Problem constants (from reference): B=16, T=512, D=1024, E=8
#define MOE_D   1024
#define MOE_E   8
#define TILE_FLOATS (MOE_D * MOE_E)     // 8192 floats = 32 KB per (b,t) tile
#define BLOCK_THREADS 256               // 8 waves (wave32)

#if defined(__has_builtin)
#  if __has_builtin(__builtin_amdgcn_global_load_async_to_lds_b128)
#    define MOE_HAVE_ASYNC_LDS 1
#  endif
#endif
#ifndef MOE_HAVE_ASYNC_LDS
#  define MOE_HAVE_ASYNC_LDS 0
#endif

// Builtin signature (from hipcc diagnostic): pointee is a 16-byte int vector,
// non-const; AS1 (global) source, AS3 (LDS) destination, imm offset, imm cpol.
typedef int v4i __attribute__((vector_size(16)));
typedef __attribute__((address_space(1))) v4i* gvec_t;
typedef __attribute__((address_space(3))) v4i* lvec_t;

__global__ __launch_bounds__(BLOCK_THREADS)
void moe_select_gather_kernel(const float* __restrict__ experts,
                              const float* __restrict__ gate,
                              float* __restrict__ out)
{
    const int bt = blockIdx.x;                       // 0 .. B*T-1
    const size_t tile_base = (size_t)bt * TILE_FLOATS;

    // ---- argmax over 8 gate logits (uniform per block; strict '>' keeps
    //      first occurrence to match jnp.argmax tie-breaking) ----
    const float* g = gate + (size_t)bt * MOE_E;
    int idx = 0;
    float m = g[0];
#pragma unroll
    for (int e = 1; e < MOE_E; ++e) {
        float v = g[e];
        if (v > m) { m = v; idx = e; }
    }

    const int lane = threadIdx.x & 31;
    const int wave = threadIdx.x >> 5;               // 0..7

#if MOE_HAVE_ASYNC_LDS
    // ================= CDNA5 async-copy path =================
    // Each wave streams its 4 KB sub-tile (128 d-values) to LDS with
    // GLOBAL_LOAD_ASYNC_TO_LDS_B128 (tracked by ASYNCcnt), then gathers the
    // selected expert lane from LDS and writes coalesced output.
    __shared__ float tile[TILE_FLOATS];              // 32 KB

    {
        char* gsrc = (char*)(experts + tile_base) +
                     (size_t)wave * 4096 + (size_t)lane * 16;
        char* lbase = (char*)tile + (size_t)wave * 4096 + (size_t)lane * 16;
#pragma unroll
        for (int c = 0; c < 8; ++c) {
            __builtin_amdgcn_global_load_async_to_lds_b128(
                (gvec_t)(gsrc + c * 512),
                (lvec_t)(lbase + c * 512),
                /*imm offset*/ 0, /*cpol*/ 0);
        }
    }
    // Wait for this wave's async copies; "memory" clobber orders the LDS reads.
    asm volatile("s_wait_asynccnt 0" ::: "memory");

#pragma unroll
    for (int k = 0; k < 4; ++k) {
        const int d = wave * 128 + k * 32 + lane;    // 0..1023, coalesced store
        out[(size_t)bt * MOE_D + d] = tile[d * MOE_E + idx];
    }
#else
    // ================= direct coalesced-load fallback =================
    // Each thread loads the full 32 B (all 8 experts) for one d with two
    // b128 loads (perfectly coalesced stream), selects via cndmask chain.
#pragma unroll
    for (int k = 0; k < 4; ++k) {
        const int d = (int)threadIdx.x + k * BLOCK_THREADS;
        const float4* p =
            reinterpret_cast<const float4*>(experts + tile_base + (size_t)d * MOE_E);
        float4 lo = p[0];
        float4 hi = p[1];
        float4 q;
        q.x = (idx & 4) ? hi.x : lo.x;
        q.y = (idx & 4) ? hi.y : lo.y;
        q.z = (idx & 4) ? hi.z : lo.z;
        q.w = (idx & 4) ? hi.w : lo.w;
        float a = (idx & 2) ? q.z : q.x;
        float b = (idx & 2) ? q.w : q.y;
        out[(size_t)bt * MOE_D + d] = (idx & 1) ? b : a;
    }
#endif
}

extern "C" void kernel_launch(void* const* d_in, const int* in_sizes, int n_in,
                              void* d_out, int out_size, void* d_ws, size_t ws_size,
                              hipStream_t stream)
{
    (void)n_in; (void)out_size; (void)d_ws; (void)ws_size;
    const float* experts = (const float*)d_in[0];   // (B,T,D,E) fp32
    const float* gate    = (const float*)d_in[1];   // (B,T,E)   fp32
    float*       out     = (float*)d_out;           // (B,T,D)   fp32

    const int bt = in_sizes[1] / MOE_E;             // B*T = 8192
    dim3 grid(bt), block(BLOCK_THREADS);
    hipLaunchKernelGGL(moe_select_gather_kernel, grid, block, 0, stream,
                       experts, gate, out);
}